// SimpleSNN_28166395528089
// MI455X (gfx1250) — compile-verified
//
#include <hip/hip_runtime.h>
#include <stdint.h>

// LIF scan over T for (B,T,N) = (256,200,1024) fp32.
// Bandwidth-bound streaming recurrence: async-LDS deep prefetch (CDNA5
// GLOBAL_LOAD_ASYNC_TO_LDS_B128 + ASYNCcnt) hides the 4KB-strided t-walk.

typedef __attribute__((ext_vector_type(4))) float f32x4;

#define B_      256
#define T_      200
#define N_      1024
#define TPB     128                 // 4 waves (wave32)
#define VEC     4                   // floats per thread per step (b128)
#define CHUNK   (TPB * VEC)         // 512 floats per block per step
#define NCHUNK  (N_ / CHUNK)        // 2
#define STAGES  16                  // async prefetch depth (ring in LDS)

#define DECAY_  0.9f
#define TH_     1.5f

__device__ __forceinline__ f32x4 lif_step(f32x4& mem, f32x4 x) {
    f32x4 m, s;
    m.x = __builtin_fmaf(mem.x, DECAY_, x.x);
    m.y = __builtin_fmaf(mem.y, DECAY_, x.y);
    m.z = __builtin_fmaf(mem.z, DECAY_, x.z);
    m.w = __builtin_fmaf(mem.w, DECAY_, x.w);
    s.x = (m.x >= TH_) ? 1.0f : 0.0f;
    s.y = (m.y >= TH_) ? 1.0f : 0.0f;
    s.z = (m.z >= TH_) ? 1.0f : 0.0f;
    s.w = (m.w >= TH_) ? 1.0f : 0.0f;
    mem.x = (m.x >= TH_) ? 0.0f : m.x;
    mem.y = (m.y >= TH_) ? 0.0f : m.y;
    mem.z = (m.z >= TH_) ? 0.0f : m.z;
    mem.w = (m.w >= TH_) ? 0.0f : m.w;
    return s;
}

__global__ __launch_bounds__(TPB) void lif_scan_async_28166395528089(
        const float* __restrict__ in, float* __restrict__ out) {
    __shared__ f32x4 stage[STAGES][TPB];   // 32 KB ring

    const int tid = threadIdx.x;
    const int b   = blockIdx.y;
    const int n0  = blockIdx.x * CHUNK + tid * VEC;

    const size_t base = (size_t)b * T_ * N_ + (size_t)n0;
    const float* gin  = in + base;

    // LDS byte address of this thread's slot-0 cell (workgroup-relative offset).
    const uint32_t lds0       = (uint32_t)(uintptr_t)(&stage[0][tid]);
    const uint32_t slot_bytes = TPB * (uint32_t)sizeof(f32x4);

    // ---- prologue: stage t = 0 .. STAGES-2 (15 async loads in flight) ----
    {
        const float* g = gin;
        #pragma unroll
        for (int s = 0; s < STAGES - 1; ++s) {
            uint32_t laddr = lds0 + (uint32_t)s * slot_bytes;
            asm volatile("global_load_async_to_lds_b128 %0, %1, off"
                         :: "v"(laddr), "v"(g) : "memory");
            g += N_;
        }
    }

    f32x4 mem = {0.0f, 0.0f, 0.0f, 0.0f};
    f32x4* outv = (f32x4*)(out + base);
    const float* gpre = gin + (size_t)(STAGES - 1) * N_;

    // ---- steady state: issue t+15, wait oldest, consume, store ----
    int t = 0;
    for (; t <= T_ - STAGES; ++t) {
        const int pre_slot = (t + STAGES - 1) & (STAGES - 1);
        uint32_t laddr = lds0 + (uint32_t)pre_slot * slot_bytes;
        asm volatile("global_load_async_to_lds_b128 %0, %1, off"
                     :: "v"(laddr), "v"(gpre) : "memory");
        gpre += N_;

        // 16 outstanding max; loads complete in order -> stage t has landed.
        asm volatile("s_wait_asynccnt 15" ::: "memory");

        f32x4 x   = stage[t & (STAGES - 1)][tid];      // ds_load_b128
        f32x4 spk = lif_step(mem, x);
        __builtin_nontemporal_store(spk, &outv[(size_t)t * (N_ / VEC)]);
    }

    // ---- tail: remaining STAGES-1 steps are fully staged ----
    asm volatile("s_wait_asynccnt 0" ::: "memory");
    for (; t < T_; ++t) {
        f32x4 x   = stage[t & (STAGES - 1)][tid];
        f32x4 spk = lif_step(mem, x);
        __builtin_nontemporal_store(spk, &outv[(size_t)t * (N_ / VEC)]);
    }
}

extern "C" void kernel_launch(void* const* d_in, const int* in_sizes, int n_in,
                              void* d_out, int out_size, void* d_ws, size_t ws_size,
                              hipStream_t stream) {
    (void)in_sizes; (void)n_in; (void)d_ws; (void)ws_size; (void)out_size;
    const float* in = (const float*)d_in[0];
    float* out      = (float*)d_out;
    dim3 grid(NCHUNK, B_, 1);      // (2, 256): 512 blocks x 4 waves
    lif_scan_async_28166395528089<<<grid, TPB, 0, stream>>>(in, out);
}